// PositionEmbeddingProj_506806141283
// MI455X (gfx1250) — compile-verified
//
#include <hip/hip_runtime.h>
#include <hip/hip_bf16.h>
#include <math.h>

typedef __attribute__((ext_vector_type(2))) float v2f;
typedef __attribute__((ext_vector_type(8))) float v8f;

#define HV    32
#define WV    88
#define HW2D  2816      // 32*88
#define EMB   256
#define CIN   512
#define DNUM  325
#define ZNUM  10
#define YX3D  32768     // 128*256
#define PE0   64

// ---------------------------------------------------------------------------
// Generic fp32 WMMA GEMM:  C[M,N] = A[M,K] @ B[K,N] + bias[M]
// Row-major everywhere. N must be a multiple of 64, K a multiple of 4.
// One wave computes a 16(M) x 64(N) strip via 4x v_wmma_f32_16x16x4_f32
// per K-step, reusing the A fragment across the 4 N-tiles.
// M-overhang rows are handled by CLAMPING the A row index (garbage rows are
// computed but never stored), keeping all loads unconditional so EXEC stays
// all-1s with no per-load exec masking.
// ---------------------------------------------------------------------------
__global__ __launch_bounds__(256) void wmma_gemm_f32(
    const float* __restrict__ A, const float* __restrict__ B,
    const float* __restrict__ bias, float* __restrict__ C,
    int M, int N, int K)
{
    const int wid  = threadIdx.x >> 5;
    const int lane = threadIdx.x & 31;
    const int mtiles  = (M + 15) >> 4;
    const int nstrips = N >> 6;
    const int wgid = blockIdx.x * 8 + wid;            // wave-uniform
    if (wgid >= mtiles * nstrips) return;             // uniform within wave -> EXEC stays all-1

    const int mt = wgid / nstrips;
    const int ns = wgid % nstrips;

    int arow = mt * 16 + (lane & 15);                 // A row for this lane
    if (arow >= M) arow = M - 1;                      // clamp (padded rows never stored)
    const int kh   = (lane >> 4) << 1;                // lanes 0-15 -> K0/K1, 16-31 -> K2/K3
    const int ncol = ns * 64 + (lane & 15);           // base N column

    const float* Ap  = A + (long)arow * K + kh;       // 8B-aligned (K%4==0, kh even)
    const float* Bp0 = B + (long)kh * N + ncol;       // row kh
    const float* Bp1 = Bp0 + N;                       // row kh+1
    const long   bstep = 4L * N;

    v8f acc0 = {}, acc1 = {}, acc2 = {}, acc3 = {};
    for (int k0 = 0; k0 < K; k0 += 4) {
        const v2f a = *(const v2f*)Ap;
        v2f b0, b1, b2, b3;
        b0.x = Bp0[0];  b0.y = Bp1[0];
        b1.x = Bp0[16]; b1.y = Bp1[16];
        b2.x = Bp0[32]; b2.y = Bp1[32];
        b3.x = Bp0[48]; b3.y = Bp1[48];

        acc0 = __builtin_amdgcn_wmma_f32_16x16x4_f32(false, a, false, b0, (short)0, acc0, false, false);
        acc1 = __builtin_amdgcn_wmma_f32_16x16x4_f32(false, a, false, b1, (short)0, acc1, false, false);
        acc2 = __builtin_amdgcn_wmma_f32_16x16x4_f32(false, a, false, b2, (short)0, acc2, false, false);
        acc3 = __builtin_amdgcn_wmma_f32_16x16x4_f32(false, a, false, b3, (short)0, acc3, false, false);

        Ap  += 4;
        Bp0 += bstep;
        Bp1 += bstep;
    }

    const int mhalf = (lane >> 4) << 3;               // lanes 16-31 hold M = r+8
#pragma unroll
    for (int r = 0; r < 8; ++r) {
        const int m = mt * 16 + r + mhalf;
        if (m < M) {
            const float bb = bias[m];
            float* Crow = C + (long)m * N + ns * 64 + (lane & 15);
            Crow[0]  = acc0[r] + bb;
            Crow[16] = acc1[r] + bb;
            Crow[32] = acc2[r] + bb;
            Crow[48] = acc3[r] + bb;
        }
    }
}

// ---------------------------------------------------------------------------
// 4x4 inverse of [ego2imgs; 0 0 0 1] (Gauss-Jordan, one thread).
// Writes rows 0..2, cols 0..3 of the inverse (12 floats).
// ---------------------------------------------------------------------------
__global__ void inv4_kernel(const float* __restrict__ ego, float* __restrict__ Minv)
{
    if (threadIdx.x != 0 || blockIdx.x != 0) return;
    float a[4][8];
    for (int r = 0; r < 3; ++r)
        for (int c = 0; c < 4; ++c) a[r][c] = ego[r * 4 + c];
    a[3][0] = 0.f; a[3][1] = 0.f; a[3][2] = 0.f; a[3][3] = 1.f;
    for (int r = 0; r < 4; ++r)
        for (int c = 0; c < 4; ++c) a[r][4 + c] = (r == c) ? 1.f : 0.f;

    for (int i = 0; i < 4; ++i) {
        int p = i;
        for (int r = i + 1; r < 4; ++r)
            if (fabsf(a[r][i]) > fabsf(a[p][i])) p = r;
        if (p != i)
            for (int c = 0; c < 8; ++c) { float t = a[i][c]; a[i][c] = a[p][c]; a[p][c] = t; }
        const float inv = 1.0f / a[i][i];
        for (int c = 0; c < 8; ++c) a[i][c] *= inv;
        for (int r = 0; r < 4; ++r) {
            if (r == i) continue;
            const float f = a[r][i];
            for (int c = 0; c < 8; ++c) a[r][c] -= f * a[i][c];
        }
    }
    for (int r = 0; r < 3; ++r)
        for (int c = 0; c < 4; ++c) Minv[r * 4 + c] = a[r][4 + c];
}

// ---------------------------------------------------------------------------
// 2D branch reduction: softmax over D of logits, expected depth, project back
// to ego frame, then red2[e, n] = w_pe0[e,:] . ptbar(n) + b_pe0[e].
// One thread per image pixel (n = h*88 + w), coalesced over logit rows.
// ---------------------------------------------------------------------------
__global__ __launch_bounds__(256) void red2_kernel(
    const float* __restrict__ logits,   // [325, 2816]
    const float* __restrict__ Minv,     // 12 floats
    const float* __restrict__ w_pe0,    // [64,3]
    const float* __restrict__ b_pe0,    // [64]
    const int* __restrict__ pad_h_p, const int* __restrict__ pad_w_p,
    float* __restrict__ red2)           // [64, 2816]
{
    const int n = blockIdx.x * blockDim.x + threadIdx.x;
    if (n >= HW2D) return;
    const int h = n / WV, w = n % WV;
    const float cw = (float)w * ((float)(*pad_w_p) / (float)WV);
    const float ch = (float)h * ((float)(*pad_h_p) / (float)HV);

    float mx = -INFINITY;
    for (int d = 0; d < DNUM; ++d) mx = fmaxf(mx, logits[d * HW2D + n]);
    float se = 0.f, sd = 0.f;
    for (int d = 0; d < DNUM; ++d) {
        const float e = __expf(logits[d * HW2D + n] - mx);
        se += e;
        sd = fmaf(e, 3.0f + 0.2f * (float)d, sd);     // coords_d = 3 + 0.2*d
    }
    const float dbar = sd / se;                       // E[depth] (softmax sums to 1)

    const float px = dbar * (Minv[0] * cw + Minv[1] * ch + Minv[2])  + Minv[3];
    const float py = dbar * (Minv[4] * cw + Minv[5] * ch + Minv[6])  + Minv[7];
    const float pz = dbar * (Minv[8] * cw + Minv[9] * ch + Minv[10]) + Minv[11];

    for (int e = 0; e < PE0; ++e)
        red2[e * HW2D + n] =
            fmaf(w_pe0[e * 3 + 0], px,
            fmaf(w_pe0[e * 3 + 1], py,
            fmaf(w_pe0[e * 3 + 2], pz, b_pe0[e])));
}

// ---------------------------------------------------------------------------
// 3D branch reduction: 10-way height softmax per BEV cell, expected z, then
// red3[e, t] = w_pe0[e,:] . (gx, gy_flipped, zbar) + b_pe0[e].
// One thread per BEV cell (t = y*256 + x), coalesced over map channels.
// ---------------------------------------------------------------------------
__global__ __launch_bounds__(256) void red3_kernel(
    const float* __restrict__ map_feat, // [256, 32768]
    const float* __restrict__ w_z,      // [10, 256]
    const float* __restrict__ b_z,      // [10]
    const float* __restrict__ w_pe0, const float* __restrict__ b_pe0,
    float* __restrict__ red3)           // [64, 32768]
{
    const int t = blockIdx.x * blockDim.x + threadIdx.x;
    if (t >= YX3D) return;

    float lg[ZNUM];
#pragma unroll
    for (int z = 0; z < ZNUM; ++z) lg[z] = b_z[z];
    for (int c = 0; c < EMB; ++c) {
        const float m = map_feat[c * YX3D + t];
#pragma unroll
        for (int z = 0; z < ZNUM; ++z) lg[z] = fmaf(w_z[z * EMB + c], m, lg[z]);
    }
    float mx = lg[0];
#pragma unroll
    for (int z = 1; z < ZNUM; ++z) mx = fmaxf(mx, lg[z]);
    float se = 0.f, sz = 0.f;
#pragma unroll
    for (int z = 0; z < ZNUM; ++z) {
        const float e = __expf(lg[z] - mx);
        se += e;
        sz = fmaf(e, -1.0f + 0.2f * (float)z, sz);    // cz = -1 + 0.2*z
    }
    const float zbar = sz / se;

    const int x = t & 255, y = t >> 8;
    const float gx = 5.0f + 0.234375f * (float)x;
    const float gy = -15.0f + 0.234375f * (float)(127 - y);  // Y-flip from reference

    for (int e = 0; e < PE0; ++e)
        red3[e * YX3D + t] =
            fmaf(w_pe0[e * 3 + 0], gx,
            fmaf(w_pe0[e * 3 + 1], gy,
            fmaf(w_pe0[e * 3 + 2], zbar, b_pe0[e])));
}

static inline int gemm_blocks(int M, int N)
{
    const int mtiles = (M + 15) / 16;
    const int nstrips = N / 64;
    return (mtiles * nstrips + 7) / 8;   // 8 waves / 256-thread block
}

extern "C" void kernel_launch(void* const* d_in, const int* in_sizes, int n_in,
                              void* d_out, int out_size, void* d_ws, size_t ws_size,
                              hipStream_t stream)
{
    (void)in_sizes; (void)n_in; (void)out_size; (void)ws_size;

    const float* img_feat = (const float*)d_in[0];
    const float* ego2imgs = (const float*)d_in[1];
    const int*   pad_h    = (const int*)d_in[2];
    const int*   pad_w    = (const int*)d_in[3];
    const float* w_in  = (const float*)d_in[4];
    const float* b_in  = (const float*)d_in[5];
    const float* w_d   = (const float*)d_in[6];
    const float* b_d   = (const float*)d_in[7];
    const float* w_z   = (const float*)d_in[8];
    const float* b_z   = (const float*)d_in[9];
    const float* w_pe0 = (const float*)d_in[10];
    const float* b_pe0 = (const float*)d_in[11];
    const float* w_pe1 = (const float*)d_in[12];
    const float* b_pe1 = (const float*)d_in[13];
    const float* map_feat = (const float*)d_in[14];

    float* out = (float*)d_out;
    float* out_e2d  = out;                                   // (256,32,88)
    float* out_e3d  = out + (EMB * HW2D);                    // (256,128,256)
    float* out_feat = out_e3d + (EMB * (size_t)YX3D);        // (256,32,88)
    float* out_map  = out_feat + (EMB * HW2D);               // (256,128,256)

    float* ws      = (float*)d_ws;
    float* Minv    = ws;                                     // 12 (use 64 slot)
    float* logits  = ws + 64;                                // 325*2816
    float* red2    = logits + (size_t)DNUM * HW2D;           // 64*2816
    float* red3    = red2 + (size_t)PE0 * HW2D;              // 64*32768

    // 1) invert ego matrix
    hipLaunchKernelGGL(inv4_kernel, dim3(1), dim3(1), 0, stream, ego2imgs, Minv);

    // 2) feat = w_in @ img_feat + b_in   (also output #3)
    hipLaunchKernelGGL(wmma_gemm_f32, dim3(gemm_blocks(EMB, HW2D)), dim3(256), 0, stream,
                       w_in, img_feat, b_in, out_feat, EMB, HW2D, CIN);

    // 3) depth logits = w_d @ feat + b_d
    hipLaunchKernelGGL(wmma_gemm_f32, dim3(gemm_blocks(DNUM, HW2D)), dim3(256), 0, stream,
                       w_d, out_feat, b_d, logits, DNUM, HW2D, EMB);

    // 4) softmax over D -> expected depth -> red2
    hipLaunchKernelGGL(red2_kernel, dim3((HW2D + 255) / 256), dim3(256), 0, stream,
                       logits, Minv, w_pe0, b_pe0, pad_h, pad_w, red2);

    // 5) embeds2d = w_pe1 @ red2 + b_pe1
    hipLaunchKernelGGL(wmma_gemm_f32, dim3(gemm_blocks(EMB, HW2D)), dim3(256), 0, stream,
                       w_pe1, red2, b_pe1, out_e2d, EMB, HW2D, PE0);

    // 6) height softmax -> expected z -> red3
    hipLaunchKernelGGL(red3_kernel, dim3(YX3D / 256), dim3(256), 0, stream,
                       map_feat, w_z, b_z, w_pe0, b_pe0, red3);

    // 7) embeds3d = w_pe1 @ red3 + b_pe1
    hipLaunchKernelGGL(wmma_gemm_f32, dim3(gemm_blocks(EMB, YX3D)), dim3(256), 0, stream,
                       w_pe1, red3, b_pe1, out_e3d, EMB, YX3D, PE0);

    // 8) map_b = map_feat (B=1 tile is a straight copy)
    hipMemcpyAsync(out_map, map_feat, (size_t)EMB * YX3D * sizeof(float),
                   hipMemcpyDeviceToDevice, stream);
}